// MSM3D_InterpolationDecoder_43980465111227
// MI455X (gfx1250) — compile-verified
//
#include <hip/hip_runtime.h>
#include <cstdint>
#include <cstddef>

// ---------------------------------------------------------------------------
// MSM3D interpolation decoder for MI455X (gfx1250, wave32).
// Memory-bound scatter+gather; no matmul structure exists (each query point
// owns a private 8xC corner matrix), so the optimization targets are HBM/L2
// data movement, not WMMA:
//   * level-1 dense grid (134 MB) < L2 (192 MB), ~15 touches/cell -> RT loads
//   * level-0 dense grid (537 MB) > L2, ~2 touches/cell -> non-temporal
//   * b128 coalesced corner fetches, 8 loads in flight before accumulate
//   * global_prefetch_b8 on the streaming coordinate reads
// Workspace layout (needs 256 + 537MB + 134MB bytes):
//   [0,256)        : sizes0 (6 ints), sizes1 (6 ints), padding
//   [256, +537MB)  : dense0  (2,128,128,128,32) f32
//   then  +134MB   : dense1  (2, 64, 64, 64,64) f32
// ---------------------------------------------------------------------------

typedef float v4f __attribute__((ext_vector_type(4)));
typedef int   v4i __attribute__((ext_vector_type(4)));

#define NB   2      // batches
#define G0   128
#define G1   64
#define C0   32
#define C1   64

// ---------- per-batch spatial extent: segment_max over idx[:,1:4] ----------
__global__ __launch_bounds__(256) void sizes_kernel(const int* __restrict__ idx,
                                                    int M,
                                                    int* __restrict__ smx /*6 ints*/) {
  __shared__ int sh[6];
  if (threadIdx.x < 6) sh[threadIdx.x] = 0;
  __syncthreads();
  int p0z = 0, p0y = 0, p0x = 0, p1z = 0, p1y = 0, p1x = 0;
  const v4i* idx4 = (const v4i*)idx;
  for (long i = (long)blockIdx.x * blockDim.x + threadIdx.x; i < M;
       i += (long)gridDim.x * blockDim.x) {
    v4i r = idx4[i];                       // [batch, z, y, x] as one b128 load
    if (r.x == 0) {
      p0z = max(p0z, r.y); p0y = max(p0y, r.z); p0x = max(p0x, r.w);
    } else {
      p1z = max(p1z, r.y); p1y = max(p1y, r.z); p1x = max(p1x, r.w);
    }
  }
  atomicMax(&sh[0], p0z); atomicMax(&sh[1], p0y); atomicMax(&sh[2], p0x);
  atomicMax(&sh[3], p1z); atomicMax(&sh[4], p1y); atomicMax(&sh[5], p1x);
  __syncthreads();
  if (threadIdx.x < 6) atomicMax(&smx[threadIdx.x], sh[threadIdx.x]);
}

// ---------- densify: one thread per (row, float4 channel chunk) ----------
template<int CVEC, int G, bool NT>
__global__ __launch_bounds__(256) void scatter_kernel(const v4f* __restrict__ feats,
                                                      const int* __restrict__ idx,
                                                      long total,
                                                      v4f* __restrict__ dense) {
  long t = (long)blockIdx.x * blockDim.x + threadIdx.x;
  if (t >= total) return;
  constexpr int SH = (CVEC == 8) ? 3 : 4;
  long m = t >> SH;
  int  q = (int)(t & (CVEC - 1));
  v4i r = ((const v4i*)idx)[m];            // [batch, z, y, x]
  size_t a = ((((size_t)r.x * G + r.y) * G + r.z) * G + r.w) * CVEC + q;
  v4f val = feats[t];
  if constexpr (NT) __builtin_nontemporal_store(val, &dense[a]);
  else              dense[a] = val;
}

// ---------- trilinear gather: one thread per (point, float4 channel chunk) --
template<int G, int CVEC, bool NT>
__global__ __launch_bounds__(256) void gather_kernel(
    const int* __restrict__ gc, const int* __restrict__ bids,
    const int* __restrict__ smx, const v4f* __restrict__ dense,
    float* __restrict__ out, int N, float invLevel, int outOff)
{
  constexpr int SH = (CVEC == 8) ? 3 : 4;
  long t = (long)blockIdx.x * blockDim.x + threadIdx.x;
  int n = (int)(t >> SH);
  int q = (int)(t & (CVEC - 1));
  if (n >= N) return;

  int b = bids[n];
  int   i0[3]; float fr[3]; int sz[3];
#pragma unroll
  for (int d = 0; d < 3; ++d) {
    int s = smx[b * 3 + d] + 1;            // per-batch spatial size
    float szf = (float)s;
    // p = (c + 0.5)/level * (sz-1)/sz   (align_corners=True mapping)
    float p = ((float)gc[3 * n + d] + 0.5f) * invLevel * (szf - 1.0f) / szf;
    float fl = floorf(p);
    i0[d] = (int)fl; fr[d] = p - fl; sz[d] = s;
  }

  size_t addr[8]; float w[8];
#pragma unroll
  for (int k = 0; k < 8; ++k) {
    size_t a = (size_t)b;
    float  ww = 1.0f;
    bool   valid = true;
#pragma unroll
    for (int d = 0; d < 3; ++d) {
      int o = (k >> (2 - d)) & 1;          // (dz,dy,dx) over dims 0,1,2
      int c = i0[d] + o;
      valid = valid && (c >= 0) && (c < sz[d]);
      ww *= o ? fr[d] : (1.0f - fr[d]);
      int cc = c < 0 ? 0 : (c > G - 1 ? G - 1 : c);
      a = a * G + cc;
    }
    addr[k] = a * CVEC + q;
    w[k] = valid ? ww : 0.0f;
  }

  // Issue all 8 corner b128 loads before any accumulation (max MLP).
  v4f v[8];
#pragma unroll
  for (int k = 0; k < 8; ++k) {
    if constexpr (NT) v[k] = __builtin_nontemporal_load(&dense[addr[k]]);
    else              v[k] = dense[addr[k]];
  }
  v4f acc = {0.0f, 0.0f, 0.0f, 0.0f};
#pragma unroll
  for (int k = 0; k < 8; ++k) acc += v[k] * w[k];

  *(v4f*)(out + (size_t)n * (C0 + C1) + outOff + 4 * q) = acc;

  // Prefetch the coordinate stream ahead (gfx1250 global_prefetch_b8).
  int pn = n + 8192;
  if (pn < N) __builtin_prefetch(gc + 3 * pn);
}

extern "C" void kernel_launch(void* const* d_in, const int* in_sizes, int n_in,
                              void* d_out, int out_size, void* d_ws, size_t ws_size,
                              hipStream_t stream) {
  (void)n_in; (void)out_size; (void)ws_size;
  const int*   gc   = (const int*)d_in[0];   // (N,3)
  const int*   bids = (const int*)d_in[1];   // (N,)
  const float* f0   = (const float*)d_in[2]; // (M0,32)
  const int*   ix0  = (const int*)d_in[3];   // (M0,4)
  const float* f1   = (const float*)d_in[4]; // (M1,64)
  const int*   ix1  = (const int*)d_in[5];   // (M1,4)

  const int  N  = in_sizes[0] / 3;
  const long M0 = in_sizes[2] / C0;
  const long M1 = in_sizes[4] / C1;

  int*   smx0 = (int*)d_ws;                  // 6 ints
  int*   smx1 = smx0 + 8;                    // 6 ints
  float* dense0 = (float*)((char*)d_ws + 256);
  const size_t n0 = (size_t)NB * G0 * G0 * G0 * C0;   // floats, 537 MB
  const size_t n1 = (size_t)NB * G1 * G1 * G1 * C1;   // floats, 134 MB
  float* dense1 = dense0 + n0;

  // Zero sizes + both dense grids (deterministic per call; ws is poisoned).
  hipMemsetAsync(d_ws, 0, 256 + (n0 + n1) * sizeof(float), stream);

  sizes_kernel<<<128, 256, 0, stream>>>(ix0, (int)M0, smx0);
  sizes_kernel<<<128, 256, 0, stream>>>(ix1, (int)M1, smx1);

  const long t0 = M0 * (C0 / 4);
  const long t1 = M1 * (C1 / 4);
  // dense0 (537 MB) streams -> NT stores keep it out of L2; dense1 stays RT.
  scatter_kernel<C0 / 4, G0, true ><<<(t0 + 255) / 256, 256, 0, stream>>>(
      (const v4f*)f0, ix0, t0, (v4f*)dense0);
  scatter_kernel<C1 / 4, G1, false><<<(t1 + 255) / 256, 256, 0, stream>>>(
      (const v4f*)f1, ix1, t1, (v4f*)dense1);

  const long g0 = (long)N * (C0 / 4);
  const long g1 = (long)N * (C1 / 4);
  // Level 0: NT loads (537 MB, ~2 touches/cell -> don't thrash L2).
  gather_kernel<G0, C0 / 4, true ><<<(g0 + 255) / 256, 256, 0, stream>>>(
      gc, bids, smx0, (const v4f*)dense0, (float*)d_out, N, 1.0f, 0);
  // Level 1: RT loads (134 MB fits the 192 MB L2, ~15 touches/cell).
  gather_kernel<G1, C1 / 4, false><<<(g1 + 255) / 256, 256, 0, stream>>>(
      gc, bids, smx1, (const v4f*)dense1, (float*)d_out, N, 0.5f, C0);
}